// VectorQuantizer_38878043964076
// MI455X (gfx1250) — compile-verified
//
#include <hip/hip_runtime.h>

typedef __attribute__((ext_vector_type(2))) float v2f;
typedef __attribute__((ext_vector_type(8))) float v8f;

#define CB_N          4096
#define TOK_D         16
#define N_ROWS        65536           // 16 * 4096
#define WAVES_PER_WG  16
#define TPB           (WAVES_PER_WG * 32)   // 512
#define N_WG          (N_ROWS / (WAVES_PER_WG * 16))  // 256
#define CHUNK_STEP    16

#define OUT_Q    0
#define OUT_LOSS 1048576
#define OUT_TOK  1048579

struct Smem {
  float cb[CB_N * TOK_D];          // permuted + swizzled codebook (WMMA-B layout)
  float ss[CB_N];                  // per-code sum of squares
  int   rowBest[WAVES_PER_WG * 16];
  float wgsum[WAVES_PER_WG];
};

// LDS float offset of logical quad q (0..3) of code j, with additive (mod 4) swizzle so
// that the 16-lane ds_load_b128 pattern (stride 64B) is bank-conflict-free.
__device__ __forceinline__ int cbOff(int j, int q) {
  return j * 16 + (((q + (j >> 2)) & 3) << 2);
}

__device__ __forceinline__ v2f mk2(float a, float b) { v2f v; v.x = a; v.y = b; return v; }

__global__ __launch_bounds__(TPB)
void vq_main(const float* __restrict__ z, const float* __restrict__ cbw,
             float* __restrict__ out, float* __restrict__ ws) {
  __shared__ Smem sm;
  const int tid = threadIdx.x;

  // ---- Stage codebook into LDS in WMMA-B layout ----------------------------
  // Logical quads per code j (k-index pairs):
  //   quad0 = (k0,k1,k4,k5)   quad1 = (k8,k9,k12,k13)   <- needed by lanes h=0
  //   quad2 = (k2,k3,k6,k7)   quad3 = (k10,k11,k14,k15) <- needed by lanes h=1
#pragma unroll
  for (int it = 0; it < CB_N / TPB; ++it) {
    const int j = tid + it * TPB;
    const float4* row = (const float4*)(cbw + j * TOK_D);
    float4 r0 = row[0], r1 = row[1], r2 = row[2], r3 = row[3];
    float ssq = r0.x*r0.x + r0.y*r0.y + r0.z*r0.z + r0.w*r0.w
              + r1.x*r1.x + r1.y*r1.y + r1.z*r1.z + r1.w*r1.w
              + r2.x*r2.x + r2.y*r2.y + r2.z*r2.z + r2.w*r2.w
              + r3.x*r3.x + r3.y*r3.y + r3.z*r3.z + r3.w*r3.w;
    *(float4*)&sm.cb[cbOff(j, 0)] = make_float4(r0.x, r0.y, r1.x, r1.y);
    *(float4*)&sm.cb[cbOff(j, 1)] = make_float4(r2.x, r2.y, r3.x, r3.y);
    *(float4*)&sm.cb[cbOff(j, 2)] = make_float4(r0.z, r0.w, r1.z, r1.w);
    *(float4*)&sm.cb[cbOff(j, 3)] = make_float4(r2.z, r2.w, r3.z, r3.w);
    sm.ss[j] = ssq;
  }
  __syncthreads();

  const int wave = tid >> 5;
  const int lane = tid & 31;
  const int m    = lane & 15;        // row (A) / column (B,C,D) within tile
  const int h    = lane >> 4;        // lane half selects K sub-pairs
  const int tile = blockIdx.x * WAVES_PER_WG + wave;
  const int rowbase = tile * 16;

  // ---- A fragments: normalized z row m, K split per lane-half --------------
  const float4* zr = (const float4*)(z + (size_t)(rowbase + m) * TOK_D);
  float4 z0 = zr[0], z1 = zr[1], z2 = zr[2], z3 = zr[3];
  float sumsq = z0.x*z0.x + z0.y*z0.y + z0.z*z0.z + z0.w*z0.w
              + z1.x*z1.x + z1.y*z1.y + z1.z*z1.z + z1.w*z1.w
              + z2.x*z2.x + z2.y*z2.y + z2.z*z2.z + z2.w*z2.w
              + z3.x*z3.x + z3.y*z3.y + z3.z*z3.z + z3.w*z3.w;
  float inv = 1.0f / fmaxf(sqrtf(sumsq), 1e-12f);
  const bool hi = (h != 0);
  // 16x16x4 f32 A layout: lanes0-15 VGPR0/1 = K0/K1, lanes16-31 = K2/K3
  v2f a0 = hi ? mk2(z0.z * inv, z0.w * inv) : mk2(z0.x * inv, z0.y * inv);
  v2f a1 = hi ? mk2(z1.z * inv, z1.w * inv) : mk2(z1.x * inv, z1.y * inv);
  v2f a2 = hi ? mk2(z2.z * inv, z2.w * inv) : mk2(z2.x * inv, z2.y * inv);
  v2f a3 = hi ? mk2(z3.z * inv, z3.w * inv) : mk2(z3.x * inv, z3.y * inv);

  // ---- Score all 4096 codes in 16-wide chunks via chained f32 WMMA ---------
  float best[8];
  int   bidx[8];
#pragma unroll
  for (int r = 0; r < 8; ++r) { best[r] = -3.402823466e38f; bidx[r] = 0; }

  for (int c0 = 0; c0 < CB_N; c0 += CHUNK_STEP) {
    const int code = c0 + m;
    const float bias = -0.5f * sm.ss[code];              // fold ||c||^2 into C
    v8f acc = {bias, bias, bias, bias, bias, bias, bias, bias};
    const float4 b01 = *(const float4*)&sm.cb[cbOff(code, 2 * h)];
    const float4 b23 = *(const float4*)&sm.cb[cbOff(code, 2 * h + 1)];
    acc = __builtin_amdgcn_wmma_f32_16x16x4_f32(false, a0, false, mk2(b01.x, b01.y), (short)0, acc, false, false);
    acc = __builtin_amdgcn_wmma_f32_16x16x4_f32(false, a1, false, mk2(b01.z, b01.w), (short)0, acc, false, false);
    acc = __builtin_amdgcn_wmma_f32_16x16x4_f32(false, a2, false, mk2(b23.x, b23.y), (short)0, acc, false, false);
    acc = __builtin_amdgcn_wmma_f32_16x16x4_f32(false, a3, false, mk2(b23.z, b23.w), (short)0, acc, false, false);
#pragma unroll
    for (int r = 0; r < 8; ++r) {
      float s = acc[r];
      bool p = s > best[r];                 // strict > keeps lowest tied index
      best[r] = p ? s : best[r];
      bidx[r] = p ? code : bidx[r];
    }
  }

  // ---- Per-row argmax reduction across the 16 column lanes -----------------
#pragma unroll
  for (int r = 0; r < 8; ++r) {
    float s = best[r];
    int   bi = bidx[r];
#pragma unroll
    for (int off = 1; off < 16; off <<= 1) {
      float s2 = __shfl_xor(s, off, 32);
      int   i2 = __shfl_xor(bi, off, 32);
      if (s2 > s || (s2 == s && i2 < bi)) { s = s2; bi = i2; }
    }
    if (m == 0) {                      // lanes 0 and 16: rows r and r+8
      const int row = r + 8 * h;
      sm.rowBest[wave * 16 + row] = bi;
      out[OUT_TOK + rowbase + row] = (float)bi;
    }
  }
  __syncthreads();

  // ---- Gather quantized rows, write straight-through output, MSE partial ---
  const int row2 = lane >> 1;          // 16 rows, 2 lanes per row
  const int half = lane & 1;           // 8 floats per lane
  const int ci = sm.rowBest[wave * 16 + row2];
  const float4* cbr = (const float4*)(cbw + (size_t)ci * TOK_D + half * 8);
  const float4* zr2 = (const float4*)(z + (size_t)(rowbase + row2) * TOK_D + half * 8);
  float4 q0 = cbr[0], q1 = cbr[1];
  float4 x0 = zr2[0], x1 = zr2[1];
  float4 d0 = make_float4(q0.x - x0.x, q0.y - x0.y, q0.z - x0.z, q0.w - x0.w);
  float4 d1 = make_float4(q1.x - x1.x, q1.y - x1.y, q1.z - x1.z, q1.w - x1.w);
  float local = d0.x*d0.x + d0.y*d0.y + d0.z*d0.z + d0.w*d0.w
              + d1.x*d1.x + d1.y*d1.y + d1.z*d1.z + d1.w*d1.w;
  float4* op = (float4*)(out + OUT_Q + (size_t)(rowbase + row2) * TOK_D + half * 8);
  op[0] = make_float4(x0.x + d0.x, x0.y + d0.y, x0.z + d0.z, x0.w + d0.w);
  op[1] = make_float4(x1.x + d1.x, x1.y + d1.y, x1.z + d1.z, x1.w + d1.w);

#pragma unroll
  for (int off = 1; off < 32; off <<= 1) local += __shfl_xor(local, off, 32);
  if (lane == 0) sm.wgsum[wave] = local;
  __syncthreads();
  if (tid == 0) {
    float s = 0.f;
#pragma unroll
    for (int w = 0; w < WAVES_PER_WG; ++w) s += sm.wgsum[w];
    ws[blockIdx.x] = s;
  }
}

__global__ void vq_finish(const float* __restrict__ ws, float* __restrict__ out) {
  if (threadIdx.x == 0 && blockIdx.x == 0) {
    float s = 0.f;
    for (int i = 0; i < N_WG; ++i) s += ws[i];
    const float mean = s * (1.0f / 1048576.0f);
    out[OUT_LOSS + 0] = 1.25f * mean;   // quantizer_loss
    out[OUT_LOSS + 1] = 0.25f * mean;   // commitment_loss
    out[OUT_LOSS + 2] = mean;           // codebook_loss
  }
}

extern "C" void kernel_launch(void* const* d_in, const int* in_sizes, int n_in,
                              void* d_out, int out_size, void* d_ws, size_t ws_size,
                              hipStream_t stream) {
  const float* z   = (const float*)d_in[0];
  const float* cbw = (const float*)d_in[1];
  float* out = (float*)d_out;
  float* ws  = (float*)d_ws;
  vq_main<<<N_WG, TPB, 0, stream>>>(z, cbw, out, ws);
  vq_finish<<<1, 32, 0, stream>>>(ws, out);
}